// DynamicHierarchicalVQ_79886391705967
// MI455X (gfx1250) — compile-verified
//
#include <hip/hip_runtime.h>
#include <hip/hip_bf16.h>

// ---------------------------------------------------------------------------
// Problem geometry (fixed by the reference)
// ---------------------------------------------------------------------------
constexpr int B_ = 4, S_ = 4096, DIM_ = 256;
constexpr int M_TOK = B_ * S_;          // 16384 tokens
constexpr int D_ = 2 * DIM_;            // 512 concat dim
constexpr int K_SYM = 4096;
constexpr int K_CON = 512;
constexpr int KSTEPS = D_ / 4;          // 128 wmma k-steps of 4

// Output layout (floats, concatenated flat in return order)
constexpr size_t Z_OUT_OFF = 0;                                   // 8,388,608 (complex interleaved)
constexpr size_t OH_S_OFF  = (size_t)M_TOK * D_;                  // 8,388,608
constexpr size_t OH_C_OFF  = OH_S_OFF + (size_t)M_TOK * K_SYM;    // 75,497,472
constexpr size_t LS_OFF    = OH_C_OFF + (size_t)M_TOK * K_CON;    // 83,886,080
constexpr size_t LC_OFF    = LS_OFF + 1;
constexpr size_t SI_OFF    = LC_OFF + 1;
constexpr size_t CI_OFF    = SI_OFF + M_TOK;
constexpr size_t CONF_OFF  = CI_OFF + M_TOK;

// Workspace layout (float element offsets into d_ws)
constexpr size_t WS_ZNORM   = 0;                       // 16384
constexpr size_t WS_SYMNORM = WS_ZNORM + M_TOK;        // 4096
constexpr size_t WS_CONNORM = WS_SYMNORM + K_SYM;      // 512
constexpr size_t WS_CDIST   = WS_CONNORM + K_CON;      // 4096
constexpr size_t WS_SD      = WS_CDIST + K_SYM;        // 16384
constexpr size_t WS_LCPART  = WS_SD + M_TOK;           // 16384
constexpr size_t WS_CBEST   = WS_LCPART + M_TOK;       // 4096 (int)
constexpr size_t WS_SI      = WS_CBEST + K_SYM;        // 16384 (int)

typedef float v2f __attribute__((ext_vector_type(2)));
typedef float v8f __attribute__((ext_vector_type(8)));

__device__ __forceinline__ v8f wmma_f32_k4(v2f a, v2f b, v8f c) {
    // D = A(16x4, f32) x B(4x16, f32) + C(16x16, f32)
    return __builtin_amdgcn_wmma_f32_16x16x4_f32(false, a, false, b, (short)0, c,
                                                 false, false);
}

// ---------------------------------------------------------------------------
// P0: zero-fill the one-hot region of d_out (float4 stores)
// ---------------------------------------------------------------------------
__global__ __launch_bounds__(256) void zero_kernel(float4* __restrict__ p, size_t n4) {
    size_t i = (size_t)blockIdx.x * 256 + threadIdx.x;
    if (i < n4) p[i] = make_float4(0.f, 0.f, 0.f, 0.f);
}

// ---------------------------------------------------------------------------
// P1: row L2-norms^2 for a [rows, 512] matrix; concat mode reads (r, im)
// ---------------------------------------------------------------------------
__global__ __launch_bounds__(256) void row_norm_kernel(const float* __restrict__ r,
                                                       const float* __restrict__ im,
                                                       const float* __restrict__ direct,
                                                       float* __restrict__ out) {
    __shared__ float red[256];
    const int row = blockIdx.x, tid = threadIdx.x;
    float v0, v1;
    if (direct) {
        v0 = direct[(size_t)row * D_ + tid];
        v1 = direct[(size_t)row * D_ + DIM_ + tid];
    } else {
        v0 = r[(size_t)row * DIM_ + tid];
        v1 = im[(size_t)row * DIM_ + tid];
    }
    red[tid] = v0 * v0 + v1 * v1;
    __syncthreads();
    for (int s = 128; s > 0; s >>= 1) {
        if (tid < s) red[tid] += red[tid + s];
        __syncthreads();
    }
    if (tid == 0) out[row] = red[0];
}

// ---------------------------------------------------------------------------
// P2/P3: WMMA distance + argmin.  Block = 256 threads = 8 waves; each wave
// owns a 16-row A tile held entirely in VGPRs (128 x v2f = 256 VGPRs, legal
// on CDNA5's 1024-VGPR wave32 file).  Each 16-code B chunk is staged once per
// block into LDS in wmma-operand-swizzled [kstep][lane] layout (conflict-free)
// and consumed by all 8 waves.  dist = ||a||^2 + ||b||^2 - 2 a.b  (f32 exact,
// matching the reference formula), running min/argmin in the 8-VGPR C/D
// layout, final cross-lane argmin via shfl_xor (offsets <= 8 stay in-half).
// ---------------------------------------------------------------------------
__global__ __launch_bounds__(256) void vq_argmin_kernel(
    const float* __restrict__ aR, const float* __restrict__ aI,   // concat mode
    const float* __restrict__ Adirect,                            // direct mode
    const float* __restrict__ Bmat, int N,
    const float* __restrict__ anorm, const float* __restrict__ bnorm,
    int* __restrict__ outIdx, float* __restrict__ outDist) {
    __shared__ float sB[16 * D_];  // 32 KB swizzled B chunk

    const int tid  = threadIdx.x;
    const int wave = tid >> 5;
    const int lane = tid & 31;
    const int l16  = lane & 15;
    const int hi   = lane >> 4;          // 0: rows 0-7 half, 1: rows 8-15 half
    const int rowBase = blockIdx.x * 128 + wave * 16;
    const int row  = rowBase + l16;
    const int dimoff = 2 * hi;           // A-operand K sub-select per half

    // ---- A tile -> registers (128 x v2f) -------------------------------
    v2f a[KSTEPS];
    if (Adirect) {
        const float* ap = Adirect + (size_t)row * D_ + dimoff;
#pragma unroll
        for (int ks = 0; ks < KSTEPS; ++ks)
            a[ks] = *(const v2f*)(ap + 4 * ks);
    } else {
#pragma unroll
        for (int ks = 0; ks < KSTEPS; ++ks) {
            const int d = 4 * ks + dimoff;
            const float* src = (d < DIM_) ? (aR + (size_t)row * DIM_ + d)
                                          : (aI + (size_t)row * DIM_ + (d - DIM_));
            a[ks] = *(const v2f*)src;
        }
    }

    // per-slot row norms: slot i holds row (rowBase + i + 8*hi)
    float an[8];
#pragma unroll
    for (int i = 0; i < 8; ++i) an[i] = anorm[rowBase + i + 8 * hi];

    float minv[8];
    int   mini[8];
#pragma unroll
    for (int i = 0; i < 8; ++i) { minv[i] = 3.4e38f; mini[i] = 0; }

    const int nChunks = N >> 4;
    for (int c = 0; c < nChunks; ++c) {
        const int nBase = c << 4;
        __syncthreads();  // previous chunk's sB reads complete
        // ---- stage B chunk (16 codes x 512) into swizzled LDS ----------
        for (int idx = tid; idx < 16 * D_; idx += 256) {
            const int n = idx >> 9;        // local code 0..15
            const int d = idx & (D_ - 1);  // dim 0..511 (coalesced global read)
            const float v = Bmat[(size_t)(nBase + n) * D_ + d];
            const int ks = d >> 2, h = (d >> 1) & 1, j = d & 1;
            sB[ks * 64 + (n + 16 * h) * 2 + j] = v;
        }
        // prefetch next chunk's rows into cache while we compute
        if (c + 1 < nChunks && tid < 16)
            __builtin_prefetch(Bmat + (size_t)(nBase + 16 + tid) * D_, 0, 0);
        __syncthreads();

        // ---- 128 wmma f32 16x16x4, 4 independent accumulation chains ---
        v8f acc0 = {}, acc1 = {}, acc2 = {}, acc3 = {};
#pragma unroll
        for (int ks = 0; ks < KSTEPS; ks += 4) {
            v2f b0 = *(const v2f*)&sB[(ks + 0) * 64 + lane * 2];
            v2f b1 = *(const v2f*)&sB[(ks + 1) * 64 + lane * 2];
            v2f b2 = *(const v2f*)&sB[(ks + 2) * 64 + lane * 2];
            v2f b3 = *(const v2f*)&sB[(ks + 3) * 64 + lane * 2];
            acc0 = wmma_f32_k4(a[ks + 0], b0, acc0);
            acc1 = wmma_f32_k4(a[ks + 1], b1, acc1);
            acc2 = wmma_f32_k4(a[ks + 2], b2, acc2);
            acc3 = wmma_f32_k4(a[ks + 3], b3, acc3);
        }
        v8f acc = (acc0 + acc1) + (acc2 + acc3);

        const int   n  = nBase + l16;     // this lane's code column
        const float bn = bnorm[n];
#pragma unroll
        for (int i = 0; i < 8; ++i) {
            const float dist = an[i] + bn - 2.0f * acc[i];
            if (dist < minv[i]) { minv[i] = dist; mini[i] = n; }  // '<' keeps first
        }
    }

    // ---- cross-lane argmin within each 16-lane half --------------------
#pragma unroll
    for (int i = 0; i < 8; ++i) {
        float v  = minv[i];
        int   ix = mini[i];
#pragma unroll
        for (int off = 8; off > 0; off >>= 1) {
            const float ov = __shfl_xor(v, off, 32);
            const int   oi = __shfl_xor(ix, off, 32);
            if (ov < v || (ov == v && oi < ix)) { v = ov; ix = oi; }
        }
        if (l16 == 0) {
            const int r = rowBase + i + 8 * hi;
            outIdx[r]  = ix;
            outDist[r] = v;
        }
    }
}

// ---------------------------------------------------------------------------
// P4: per-token finalize: complex-interleaved z_out gather, one-hot scatter,
//     si/ci/conf, and lc partials (cdist[si[t]]).
// ---------------------------------------------------------------------------
__global__ __launch_bounds__(256) void finalize_kernel(
    const float* __restrict__ sym,
    const int* __restrict__ si, const float* __restrict__ sd,
    const int* __restrict__ cbest, const float* __restrict__ cdist,
    float* __restrict__ out, float* __restrict__ lcpart) {
    const int t = blockIdx.x;
    const int d = threadIdx.x;
    const int s = si[t];
    const float re = sym[(size_t)s * D_ + d];
    const float im = sym[(size_t)s * D_ + DIM_ + d];
    ((float2*)(out + Z_OUT_OFF))[(size_t)t * DIM_ + d] = make_float2(re, im);
    if (d == 0) {
        const int c = cbest[s];
        out[OH_S_OFF + (size_t)t * K_SYM + s] = 1.0f;
        out[OH_C_OFF + (size_t)t * K_CON + c] = 1.0f;
        out[SI_OFF + t]   = (float)s;
        out[CI_OFF + t]   = (float)c;
        out[CONF_OFF + t] = 1.0f / (1.0f + sd[t]);
        lcpart[t] = cdist[s];
    }
}

// ---------------------------------------------------------------------------
// P5: deterministic fixed-order loss reduction (no atomics)
// ---------------------------------------------------------------------------
__global__ __launch_bounds__(256) void loss_kernel(const float* __restrict__ sd,
                                                   const float* __restrict__ lcpart,
                                                   float* __restrict__ out) {
    __shared__ float s1[256], s2[256];
    const int tid = threadIdx.x;
    float A = 0.f, B = 0.f;
    for (int i = tid; i < M_TOK; i += 256) { A += sd[i]; B += lcpart[i]; }
    s1[tid] = A; s2[tid] = B;
    __syncthreads();
    for (int s = 128; s > 0; s >>= 1) {
        if (tid < s) { s1[tid] += s1[tid + s]; s2[tid] += s2[tid + s]; }
        __syncthreads();
    }
    if (tid == 0) {
        const float scale = 1.25f / ((float)M_TOK * (float)D_);
        out[LS_OFF] = s1[0] * scale;
        out[LC_OFF] = s2[0] * scale;
    }
}

// ---------------------------------------------------------------------------
extern "C" void kernel_launch(void* const* d_in, const int* in_sizes, int n_in,
                              void* d_out, int out_size, void* d_ws, size_t ws_size,
                              hipStream_t stream) {
    const float* z_real = (const float*)d_in[0];
    const float* z_imag = (const float*)d_in[1];
    const float* sym    = (const float*)d_in[2];
    const float* con    = (const float*)d_in[3];
    float* out = (float*)d_out;
    float* W   = (float*)d_ws;

    float* znorm   = W + WS_ZNORM;
    float* symnorm = W + WS_SYMNORM;
    float* connorm = W + WS_CONNORM;
    float* cdist   = W + WS_CDIST;
    float* sdarr   = W + WS_SD;
    float* lcpart  = W + WS_LCPART;
    int*   cbest   = (int*)(W + WS_CBEST);
    int*   siarr   = (int*)(W + WS_SI);

    // P0: zero one-hot region (oh_s + oh_c contiguous)
    const size_t n4 = ((size_t)M_TOK * K_SYM + (size_t)M_TOK * K_CON) / 4;  // 18,874,368
    zero_kernel<<<(unsigned)((n4 + 255) / 256), 256, 0, stream>>>(
        (float4*)(out + OH_S_OFF), n4);

    // P1: norms
    row_norm_kernel<<<M_TOK, 256, 0, stream>>>(z_real, z_imag, nullptr, znorm);
    row_norm_kernel<<<K_SYM, 256, 0, stream>>>(nullptr, nullptr, sym, symnorm);
    row_norm_kernel<<<K_CON, 256, 0, stream>>>(nullptr, nullptr, con, connorm);

    // P2: sym -> con nearest table (stage-2 collapses onto codebook rows)
    vq_argmin_kernel<<<K_SYM / 128, 256, 0, stream>>>(
        nullptr, nullptr, sym, con, K_CON, symnorm, connorm, cbest, cdist);

    // P3: token -> sym nearest (the big WMMA GEMM-argmin)
    vq_argmin_kernel<<<M_TOK / 128, 256, 0, stream>>>(
        z_real, z_imag, nullptr, sym, K_SYM, znorm, symnorm, siarr, sdarr);

    // P4: gather/scatter outputs
    finalize_kernel<<<M_TOK, 256, 0, stream>>>(sym, siarr, sdarr, cbest, cdist,
                                               out, lcpart);

    // P5: losses (deterministic)
    loss_kernel<<<1, 256, 0, stream>>>(sdarr, lcpart, out);
}